// PositionalCharacterLevelWordSparse_17334488007262
// MI455X (gfx1250) — compile-verified
//
#include <hip/hip_runtime.h>
#include <stdint.h>

// Problem constants (from the reference)
#define NUM_EMB   512
#define MAX_POS   16
#define DVEC      528   // NUM_EMB + MAX_POS
#define LCHARS    16
#define WAVES_PER_BLOCK 8
#define BLOCK_THREADS   (WAVES_PER_BLOCK * 32)

// One wave32 per output row. Build the 528-bin histogram in LDS, then stream
// it to global memory with gfx1250 async LDS->global b128 stores.
__global__ __launch_bounds__(BLOCK_THREADS)
void pos_char_hist_kernel(const int* __restrict__ tok,
                          const int* __restrict__ pos,
                          float* __restrict__ out,
                          int nrows) {
    __shared__ __align__(16) float lds[WAVES_PER_BLOCK * DVEC];

    const int lane = threadIdx.x & 31;
    const int wave = threadIdx.x >> 5;
    const int row  = blockIdx.x * WAVES_PER_BLOCK + wave;
    if (row >= nrows) return;   // wave-uniform

    float* lrow = &lds[wave * DVEC];

    // ---- Phase 1: zero this wave's LDS row (132 x float4 = 2112 B) ----
    float4 z; z.x = 0.f; z.y = 0.f; z.z = 0.f; z.w = 0.f;
    float4* l4 = reinterpret_cast<float4*>(lrow);
#pragma unroll
    for (int i = 0; i < 4; ++i)
        l4[lane + 32 * i] = z;          // ds_store_b128
    if (lane < 4)
        l4[128 + lane] = z;

    // ---- Phase 2: masked scatter-add into LDS ----
    // lanes 0..15 handle token bins, lanes 16..31 handle position bins.
    // Same-wave DS ops are in-order, so no barrier is needed.
    const long base = (long)row * LCHARS;
    const int  c    = lane & 15;
    const int  t    = tok[base + c];
    if (t != 0) {  // PADDING_IDX == 0 masks both histograms
        const int bin = (lane < 16) ? t : (NUM_EMB + pos[base + c]);
        atomicAdd(&lrow[bin], 1.0f);    // ds_add_f32 (no return)
    }

    // ---- Phase 3: async LDS -> global store of the finished row ----
    // DS and ASYNC use different counters: the async engine must observe the
    // completed LDS writes, so drain DScnt first.
    asm volatile("s_wait_dscnt 0x0" ::: "memory");

    const uint64_t gbase = (uint64_t)(uintptr_t)(out + (size_t)row * DVEC);
    // Low 32 bits of a generic pointer into __shared__ are the LDS byte offset.
    const unsigned lbase = (unsigned)(uintptr_t)lrow;

#pragma unroll
    for (int i = 0; i < 4; ++i) {
        const unsigned off = (unsigned)(lane + 32 * i) * 16u;
        const uint64_t ga  = gbase + off;
        const unsigned la  = lbase + off;
        asm volatile("global_store_async_from_lds_b128 %0, %1, off"
                     :: "v"(ga), "v"(la) : "memory");
    }
    if (lane < 4) {
        const unsigned off = (unsigned)(128 + lane) * 16u;
        const uint64_t ga  = gbase + off;
        const unsigned la  = lbase + off;
        asm volatile("global_store_async_from_lds_b128 %0, %1, off"
                     :: "v"(ga), "v"(la) : "memory");
    }

    // Drain ASYNCcnt before the wave retires (S_ENDPGM would also wait, but
    // be explicit so LDS can't be reallocated under pending async reads).
    asm volatile("s_wait_asynccnt 0x0" ::: "memory");
}

extern "C" void kernel_launch(void* const* d_in, const int* in_sizes, int n_in,
                              void* d_out, int out_size, void* d_ws, size_t ws_size,
                              hipStream_t stream) {
    const int* tok = (const int*)d_in[0];   // (B, W, L) int32
    const int* pos = (const int*)d_in[1];   // (B, W, L) int32
    float*     out = (float*)d_out;         // (B, W, 528) float32

    const int nrows  = in_sizes[0] / LCHARS;                 // B*W = 65536
    const int blocks = (nrows + WAVES_PER_BLOCK - 1) / WAVES_PER_BLOCK;

    pos_char_hist_kernel<<<blocks, BLOCK_THREADS, 0, stream>>>(tok, pos, out, nrows);
}